// SRenderY_29360396436096
// MI455X (gfx1250) — compile-verified
//
#include <hip/hip_runtime.h>
#include <hip/hip_bf16.h>
#include <stdint.h>

// Problem constants (from the reference): N=16, H=512, W=512, K=1, F=9976, D=12
#define N_   16
#define H_   512
#define W_   512
#define D_   12
#define HW_  (H_ * W_)              // 262144
#define PIX_TOTAL (N_ * HW_)        // 4194304 pixels (K == 1)

#define TILE      256               // pixels per tile == block size (8 waves, wave32)
#define TPB_TILES 16                // tiles per block (software pipeline depth)
#define NBLOCKS   (PIX_TOTAL / (TILE * TPB_TILES))   // 1024

typedef uint32_t v4u __attribute__((ext_vector_type(4)));
typedef int      v8i __attribute__((ext_vector_type(8)));
typedef int      v4i __attribute__((ext_vector_type(4)));

#define HAVE_TDM __has_builtin(__builtin_amdgcn_tensor_load_to_lds)

// ---------------------------------------------------------------------------
// Issue one TDM 1-D (tile = nelem x 1 dwords) copy: global -> LDS.
// Descriptor layout per CDNA5 ISA.
//   group0: [0]=count=1, [1]=lds_addr(bytes), [2]=gaddr[31:0],
//           [3]=gaddr[56:32] | type(2)<<30
//   group1: [0]=data_size(4B)<<16, tensor_dim0=nelem, tensor_dim1=1,
//           tile_dim0=nelem, tile_dim1=1, tensor_dim0_stride=nelem
// ---------------------------------------------------------------------------
__device__ __forceinline__ void tdm_load_1d(uint32_t lds_byte_off,
                                            const void* gsrc,
                                            uint32_t nelem_dw) {
#if HAVE_TDM
  const uint64_t ga = (uint64_t)(uintptr_t)gsrc;
  v4u g0;
  g0[0] = 1u;                                           // count = 1 valid descriptor
  g0[1] = lds_byte_off;                                 // LDS destination (bytes)
  g0[2] = (uint32_t)ga;                                 // global_addr[31:0]
  g0[3] = (uint32_t)((ga >> 32) & 0x1FFFFFFull)         // global_addr[56:32]
          | 0x80000000u;                                // type = 2 ("image")
  v8i g1;
  g1[0] = (int)(2u << 16);                              // data_size = 2 (4 bytes)
  g1[1] = (int)((nelem_dw & 0xFFFFu) << 16);            // tensor_dim0[15:0]
  g1[2] = (int)((nelem_dw >> 16) | (1u << 16));         // tensor_dim0[31:16] | tensor_dim1=1
  g1[3] = (int)((nelem_dw & 0xFFFFu) << 16);            // tile_dim0 = nelem
  g1[4] = 1;                                            // tile_dim1 = 1, tile_dim2 = 0
  g1[5] = (int)nelem_dw;                                // tensor_dim0_stride[31:0]
  g1[6] = 0;
  g1[7] = 0;
  const v4i z4 = {0, 0, 0, 0};
#if defined(__clang_major__) && (__clang_major__ >= 23)
  const v8i z8 = {0, 0, 0, 0, 0, 0, 0, 0};
  __builtin_amdgcn_tensor_load_to_lds(g0, g1, z4, z4, z8, 0);
#else
  __builtin_amdgcn_tensor_load_to_lds(g0, g1, z4, z4, 0);
#endif
#else
  (void)lds_byte_off; (void)gsrc; (void)nelem_dw;
#endif
}

// ---------------------------------------------------------------------------
// Interpolation kernel: one thread per pixel, TDM-staged pix/bary tiles,
// random attribute gathers via global_load_b128, NT stores of the output.
// ---------------------------------------------------------------------------
__global__ __launch_bounds__(TILE) void SRenderY_interp_kernel(
    const int*    __restrict__ pix,    // [N,H,W,1]        face index, -1 = bg
    const float*  __restrict__ bary,   // [N,H,W,1,3]      barycentric weights
    const float4* __restrict__ attr4,  // [N*F, 3, 12] f32 -> 9 float4 per face
    float*        __restrict__ out)    // [N, 13, H, W]
{
  extern __shared__ char smem[];
  int*   s_pix  = (int*)smem;                      // [2][TILE]
  float* s_bary = (float*)(smem + 2 * TILE * 4);   // [2][TILE*3]
  const uint32_t PIX_OFF[2]  = {0u, TILE * 4u};
  const uint32_t BARY_OFF[2] = {2u * TILE * 4u, 2u * TILE * 4u + TILE * 12u};

  const unsigned tid   = threadIdx.x;
  const unsigned tile0 = blockIdx.x * TPB_TILES;

#if HAVE_TDM
  // Prologue: prefetch tile 0 into buffer 0 (one issue per block: EXEC is
  // ignored by TDM, so gate on a single lane / single wave).
  if (tid == 0) {
    tdm_load_1d(PIX_OFF[0],  pix  + (size_t)tile0 * TILE,     TILE);
    tdm_load_1d(BARY_OFF[0], bary + (size_t)tile0 * TILE * 3, TILE * 3);
  }
#endif

  for (unsigned t = 0; t < TPB_TILES; ++t) {
    const unsigned buf       = t & 1u;
    const size_t   tileStart = (size_t)(tile0 + t) * TILE;

    int   p;
    float b0, b1, b2;

#if HAVE_TDM
    if (tid == 0) {
      if (t + 1 < TPB_TILES) {
        const unsigned nb = (t + 1) & 1u;
        const size_t   ns = (size_t)(tile0 + t + 1) * TILE;
        tdm_load_1d(PIX_OFF[nb],  pix  + ns,     TILE);
        tdm_load_1d(BARY_OFF[nb], bary + ns * 3, TILE * 3);
        // 2 newest ops (next tile) may remain in flight; current tile done.
        __builtin_amdgcn_s_wait_tensorcnt(2);
      } else {
        __builtin_amdgcn_s_wait_tensorcnt(0);
      }
    }
    __syncthreads();                 // TDM data visible to all 8 waves
    p  = s_pix [buf * TILE + tid];
    b0 = s_bary[buf * TILE * 3 + 3 * tid + 0];
    b1 = s_bary[buf * TILE * 3 + 3 * tid + 1];
    b2 = s_bary[buf * TILE * 3 + 3 * tid + 2];
#else
    const size_t i0 = tileStart + tid;
    p  = __builtin_nontemporal_load(pix + i0);
    b0 = __builtin_nontemporal_load(bary + i0 * 3 + 0);
    b1 = __builtin_nontemporal_load(bary + i0 * 3 + 1);
    b2 = __builtin_nontemporal_load(bary + i0 * 3 + 2);
#endif

    const bool  bg  = (p < 0);
    const int   idx = bg ? 0 : p;                 // clamp -1 -> 0 like reference
    const float bw[3] = {b0, b1, b2};

    // Gather this face's 3x12 attribute block: 9 aligned 16B loads (144 B).
    const float4* a = attr4 + (size_t)idx * 9;
    float acc[12];
#pragma unroll
    for (int j = 0; j < 12; ++j) acc[j] = 0.0f;
#pragma unroll
    for (int v = 0; v < 3; ++v) {
#pragma unroll
      for (int j = 0; j < 3; ++j) {
        const float4 A = a[v * 3 + j];
        acc[j * 4 + 0] = fmaf(bw[v], A.x, acc[j * 4 + 0]);
        acc[j * 4 + 1] = fmaf(bw[v], A.y, acc[j * 4 + 1]);
        acc[j * 4 + 2] = fmaf(bw[v], A.z, acc[j * 4 + 2]);
        acc[j * 4 + 3] = fmaf(bw[v], A.w, acc[j * 4 + 3]);
      }
    }

    const float  m  = bg ? 0.0f : 1.0f;           // zero background + vismask
    const size_t i  = tileStart + tid;
    const size_t n  = i >> 18;                    // / (H*W)
    const size_t hw = i & (size_t)(HW_ - 1);      // % (H*W)
    float* ob = out + n * (size_t)(D_ + 1) * HW_ + hw;

    // Channel-major output: 13 coalesced NT stores (write-once 218 MB stream,
    // keep L2 for the attribute table).
#pragma unroll
    for (int j = 0; j < 12; ++j)
      __builtin_nontemporal_store(m * acc[j], ob + (size_t)j * HW_);
    __builtin_nontemporal_store(m, ob + (size_t)12 * HW_);

#if HAVE_TDM
    __syncthreads();                 // buffer free before wave0 refills it
#endif
  }
}

// ---------------------------------------------------------------------------
extern "C" void kernel_launch(void* const* d_in, const int* in_sizes, int n_in,
                              void* d_out, int out_size, void* d_ws, size_t ws_size,
                              hipStream_t stream) {
  (void)in_sizes; (void)n_in; (void)out_size; (void)d_ws; (void)ws_size;
  const int*    pix   = (const int*)d_in[0];     // pix_to_face
  const float*  bary  = (const float*)d_in[1];   // bary_coords
  const float4* attr4 = (const float4*)d_in[2];  // attributes (16B-aligned rows)
  float*        out   = (float*)d_out;

  const size_t shmem = (size_t)(2 * TILE * 4 + 2 * TILE * 12);  // 8 KB
  SRenderY_interp_kernel<<<dim3(NBLOCKS), dim3(TILE), shmem, stream>>>(
      pix, bary, attr4, out);
}